// SensorCalibration_75471165325585
// MI455X (gfx1250) — compile-verified
//
#include <hip/hip_runtime.h>
#include <hip/hip_bf16.h>

// ---------------------------------------------------------------------------
// SensorCalibration on gfx1250.
//
// Phase 1 (WMMA): per sensor n, compute sufficient statistics over T=16384:
//   N9[e1][e2] = sum_t a_t[e1]*a_t[e2]   (a = vec(orgE[t,n]))   -> A^T A GEMM
//   M9[e1][e2] = sum_t a_t[e1]*b_t[e2]   (b = vec(trgE[t,n]))   -> A^T B GEMM
// via V_WMMA_F32_16X16X4_F32, K accumulated 4 timesteps per instruction.
// 17 sensors x 32 chunks = 544 waves; partial 9x9 tiles to d_ws.
//
// Phase 2: deterministic chunk reduction, build symmetrized 4-index tensors,
// then 17 threads run the exact 450-step Adam loop in closed form:
//   Q = C C^T,  H[ij] = sum_ab (N[i][a][j][b]+N[a][i][b][j]) Q[ab]
//   T[ij] = sum_pq (M[i][p][j][q]+M[p][i][q][j]) C[pq]
//   G = 2*lw*(H*C - T);  chain through Rodrigues analytically.
// ang0 reproduced with JAX Threefry-2x32 (key (0,1)).
// ---------------------------------------------------------------------------

typedef __attribute__((ext_vector_type(2))) float v2f;
typedef __attribute__((ext_vector_type(8))) float v8f;

#define T_TOTAL 16384
#define NSENS   17
#define NCHUNK  32
#define T_PER   (T_TOTAL / NCHUNK)   // 512
#define ROWSTRIDE (NSENS * 9)        // 153 floats per timestep row

// ---------------- Phase 1: WMMA statistics ----------------
__global__ void __launch_bounds__(32)
calib_stats(const float* __restrict__ org, const float* __restrict__ trg,
            float* __restrict__ part)
{
  const int wid    = blockIdx.x;
  const int sensor = wid / NCHUNK;
  const int chunk  = wid % NCHUNK;
  const int lane   = threadIdx.x;
  const int half   = lane >> 4;        // K half select
  const int col    = lane & 15;        // M (A-op) == N (B-op) column
  const int c9     = (col < 9) ? col : 0;   // pad duplicate, stays in-bounds

  size_t base = (size_t)(chunk * T_PER + 2 * half) * ROWSTRIDE + sensor * 9 + c9;
  const float* pa = org + base;
  const float* pb = trg + base;

  v8f accN = {0.f,0.f,0.f,0.f,0.f,0.f,0.f,0.f};
  v8f accM = {0.f,0.f,0.f,0.f,0.f,0.f,0.f,0.f};

  #pragma unroll 4
  for (int s = 0; s < T_PER / 4; ++s) {
    v2f a, b;
    a.x = pa[0];          // t0 + 2*half
    a.y = pa[ROWSTRIDE];  // t0 + 2*half + 1
    b.x = pb[0];
    b.y = pb[ROWSTRIDE];
    // A-operand (16x4) and B-operand (4x16) lane layouts coincide for our
    // slab, so 'a' serves as both. D += A*B accumulates a_t (outer) a_t / b_t.
    accN = __builtin_amdgcn_wmma_f32_16x16x4_f32(false, a, false, a,
                                                 (short)0, accN, false, false);
    accM = __builtin_amdgcn_wmma_f32_16x16x4_f32(false, a, false, b,
                                                 (short)0, accM, false, false);
    pa += 4 * ROWSTRIDE;
    pb += 4 * ROWSTRIDE;
  }

  // D layout: VGPR g, lane L -> row g + 8*(L>=16), col L%16.
  float* o = part + (size_t)(sensor * NCHUNK + chunk) * 162;
  if (col < 9) {
    if (half == 0) {
      #pragma unroll
      for (int g = 0; g < 8; ++g) {        // rows 0..7
        o[g * 9 + col]      = accN[g];
        o[81 + g * 9 + col] = accM[g];
      }
    } else {                                // row 8 lives in g=0, lanes 16..24
      o[72 + col]      = accN[0];
      o[81 + 72 + col] = accM[0];
    }
  }
}

// ---------------- Phase 2: reduce + Adam ----------------
__device__ __forceinline__ unsigned rotl32(unsigned x, int r) {
  return (x << r) | (x >> (32 - r));
}

__global__ void __launch_bounds__(128)
calib_adam(const float* __restrict__ part, float* __restrict__ out)
{
  __shared__ float rawN[NSENS][81];
  __shared__ float rawM[NSENS][81];
  __shared__ float NSs[NSENS][81];   // [(a,b)*9 + (i,j)]
  __shared__ float MSs[NSENS][81];
  const int tid = threadIdx.x;

  // Deterministic reduction over chunks (fixed order).
  for (int idx = tid; idx < NSENS * 81; idx += blockDim.x) {
    int n = idx / 81, e = idx - n * 81;
    const float* b = part + (size_t)n * NCHUNK * 162;
    float sN = 0.f, sM = 0.f;
    for (int c = 0; c < NCHUNK; ++c) {
      sN += b[c * 162 + e];
      sM += b[c * 162 + 81 + e];
    }
    rawN[n][e] = sN;
    rawM[n][e] = sM;
  }
  __syncthreads();

  // Symmetrized tensors: NS[(a,b)][(i,j)] = N9[3i+a][3j+b] + N9[3a+i][3b+j]
  for (int idx = tid; idx < NSENS * 81; idx += blockDim.x) {
    int n = idx / 81, e = idx - n * 81;
    int ab = e / 9, ij = e - ab * 9;
    int a = ab / 3, bb = ab - a * 3, i = ij / 3, j = ij - i * 3;
    NSs[n][e] = rawN[n][(3*i+a)*9 + (3*j+bb)] + rawN[n][(3*a+i)*9 + (3*bb+j)];
    MSs[n][e] = rawM[n][(3*i+a)*9 + (3*j+bb)] + rawM[n][(3*a+i)*9 + (3*bb+j)];
  }
  __syncthreads();

  if (tid < NSENS) {
    // ---- ang0 = jax.random.uniform(key(1), (17,3)) via Threefry-2x32 ----
    float r[3];
    #pragma unroll
    for (int comp = 0; comp < 3; ++comp) {
      int f = tid * 3 + comp;                 // flat index 0..50
      int p = (f < 26) ? f : f - 26;          // lane pair index
      unsigned x0 = (unsigned)p;              // counts[0:26]
      unsigned x1 = (p < 25) ? (unsigned)(26 + p) : 0u;  // counts[26:51]+pad
      const unsigned ks0 = 0u, ks1 = 1u, ks2 = 0x1BD11BDAu ^ 0u ^ 1u;
      const int ra[4] = {13, 15, 26, 6};
      const int rb[4] = {17, 29, 16, 24};
      x0 += ks0; x1 += ks1;
      #pragma unroll
      for (int q = 0; q < 4; ++q) { x0 += x1; x1 = rotl32(x1, ra[q]); x1 ^= x0; }
      x0 += ks1; x1 += ks2 + 1u;
      #pragma unroll
      for (int q = 0; q < 4; ++q) { x0 += x1; x1 = rotl32(x1, rb[q]); x1 ^= x0; }
      x0 += ks2; x1 += ks0 + 2u;
      #pragma unroll
      for (int q = 0; q < 4; ++q) { x0 += x1; x1 = rotl32(x1, ra[q]); x1 ^= x0; }
      x0 += ks0; x1 += ks1 + 3u;
      #pragma unroll
      for (int q = 0; q < 4; ++q) { x0 += x1; x1 = rotl32(x1, rb[q]); x1 ^= x0; }
      x0 += ks1; x1 += ks2 + 4u;
      #pragma unroll
      for (int q = 0; q < 4; ++q) { x0 += x1; x1 = rotl32(x1, ra[q]); x1 ^= x0; }
      x0 += ks2; x1 += ks0 + 5u;
      unsigned bits = (f < 26) ? x0 : x1;
      r[comp] = __uint_as_float((bits >> 9) | 0x3f800000u) - 1.0f;
    }

    const float* NSl = &NSs[tid][0];
    const float* MSl = &MSs[tid][0];
    float m[3] = {0.f, 0.f, 0.f}, v[3] = {0.f, 0.f, 0.f};
    double b1p = 1.0, b2p = 1.0;
    const float lw2 = 2.0f * 10.0f;   // 2 * lw

    for (int it = 0; it < 450; ++it) {
      float lr = (it < 150) ? 0.01f : ((it < 300) ? 0.009f : 0.0081f);

      // Rodrigues forward
      float rr  = r[0]*r[0] + r[1]*r[1] + r[2]*r[2];
      float th  = sqrtf(rr);
      float ths = fmaxf(th, 1e-12f);
      float inv = 1.0f / ths;
      float k0 = r[0]*inv, k1 = r[1]*inv, k2 = r[2]*inv;
      float cth = cosf(th), sth = sinf(th);
      float omc = 1.0f - cth;
      float C[9];
      C[0] =  cth + omc*k0*k0; C[1] = -sth*k2 + omc*k0*k1; C[2] =  sth*k1 + omc*k0*k2;
      C[3] =  sth*k2 + omc*k1*k0; C[4] =  cth + omc*k1*k1; C[5] = -sth*k0 + omc*k1*k2;
      C[6] = -sth*k1 + omc*k2*k0; C[7] =  sth*k0 + omc*k2*k1; C[8] =  cth + omc*k2*k2;

      // Q = C C^T
      float Q[9];
      #pragma unroll
      for (int i2 = 0; i2 < 3; ++i2)
        #pragma unroll
        for (int j2 = 0; j2 < 3; ++j2)
          Q[i2*3+j2] = C[i2*3+0]*C[j2*3+0] + C[i2*3+1]*C[j2*3+1] + C[i2*3+2]*C[j2*3+2];

      // H[ij] = sum_ab NS[ab][ij]*Q[ab];  T[ij] = sum_ab MS[ab][ij]*C[ab]
      float H[9], Tm[9];
      #pragma unroll
      for (int ij2 = 0; ij2 < 9; ++ij2) {
        float h = 0.f, t2 = 0.f;
        #pragma unroll
        for (int ab2 = 0; ab2 < 9; ++ab2) {
          h  = fmaf(NSl[ab2*9 + ij2], Q[ab2], h);
          t2 = fmaf(MSl[ab2*9 + ij2], C[ab2], t2);
        }
        H[ij2] = h; Tm[ij2] = t2;
      }

      // G = dL/dC = 2*lw*(H*C - T)
      float G[9];
      #pragma unroll
      for (int i2 = 0; i2 < 3; ++i2)
        #pragma unroll
        for (int j2 = 0; j2 < 3; ++j2) {
          float hc = H[i2*3+0]*C[0*3+j2] + H[i2*3+1]*C[1*3+j2] + H[i2*3+2]*C[2*3+j2];
          G[i2*3+j2] = lw2 * (hc - Tm[i2*3+j2]);
        }

      // Chain through Rodrigues: g = alpha*k + (z - (z.k)k)/theta
      float trG = G[0] + G[4] + G[8];
      float w0 = G[7] - G[5], w1 = G[2] - G[6], w2 = G[3] - G[1];   // vee(G-G^T)
      float u0 = 2.f*G[0]*k0 + (G[1]+G[3])*k1 + (G[2]+G[6])*k2;     // (G+G^T)k
      float u1 = (G[1]+G[3])*k0 + 2.f*G[4]*k1 + (G[5]+G[7])*k2;
      float u2 = (G[2]+G[6])*k0 + (G[5]+G[7])*k1 + 2.f*G[8]*k2;
      float wk  = w0*k0 + w1*k1 + w2*k2;
      float kGk = k0*(G[0]*k0 + G[1]*k1 + G[2]*k2)
                + k1*(G[3]*k0 + G[4]*k1 + G[5]*k2)
                + k2*(G[6]*k0 + G[7]*k1 + G[8]*k2);
      float alpha = -sth*trG + cth*wk + sth*kGk;
      float z0 = sth*w0 + omc*u0, z1 = sth*w1 + omc*u1, z2 = sth*w2 + omc*u2;
      float zk = z0*k0 + z1*k1 + z2*k2;
      float g0 = alpha*k0 + (z0 - zk*k0)*inv;
      float g1 = alpha*k1 + (z1 - zk*k1)*inv;
      float g2 = alpha*k2 + (z2 - zk*k2)*inv;

      // Adam (B1=0.9, B2=0.999, EPS=1e-8), bias powers in double
      b1p *= 0.9; b2p *= 0.999;
      float bc1 = (float)(1.0 - b1p), bc2 = (float)(1.0 - b2p);
      float g[3] = {g0, g1, g2};
      #pragma unroll
      for (int comp = 0; comp < 3; ++comp) {
        m[comp] = 0.9f * m[comp] + 0.1f * g[comp];
        v[comp] = 0.999f * v[comp] + 0.001f * g[comp] * g[comp];
        float mhat = m[comp] / bc1;
        float vhat = v[comp] / bc2;
        r[comp] -= lr * mhat / (sqrtf(vhat) + 1e-8f);
      }
    }

    out[tid*3 + 0] = r[0];
    out[tid*3 + 1] = r[1];
    out[tid*3 + 2] = r[2];
  }
}

extern "C" void kernel_launch(void* const* d_in, const int* in_sizes, int n_in,
                              void* d_out, int out_size, void* d_ws, size_t ws_size,
                              hipStream_t stream) {
  const float* org = (const float*)d_in[0];   // (16384,17,3,3) f32
  const float* trg = (const float*)d_in[1];   // (16384,17,3,3) f32
  float* part = (float*)d_ws;                 // 17*32*162 floats = 344 KiB

  calib_stats<<<dim3(NSENS * NCHUNK), dim3(32), 0, stream>>>(org, trg, part);
  calib_adam<<<dim3(1), dim3(128), 0, stream>>>(part, (float*)d_out);
}